// WeightedAttentionPooling_73134703116341
// MI455X (gfx1250) — compile-verified
//
#include <hip/hip_runtime.h>
#include <hip/hip_bf16.h>

typedef float v2f __attribute__((ext_vector_type(2)));
typedef float v8f __attribute__((ext_vector_type(8)));

#define DIM 128

__device__ __forceinline__ float wave_reduce_sum(float v) {
    v += __shfl_xor(v, 16, 32);
    v += __shfl_xor(v, 8, 32);
    v += __shfl_xor(v, 4, 32);
    v += __shfl_xor(v, 2, 32);
    v += __shfl_xor(v, 1, 32);
    return v;
}

// lower_bound over sorted int32 index array
__device__ __forceinline__ int lower_bound(const int* __restrict__ idx, int n, int key) {
    int lo = 0, hi = n;
    while (lo < hi) {
        int mid = (lo + hi) >> 1;
        if (idx[mid] < key) lo = mid + 1; else hi = mid;
    }
    return lo;
}

// Kernel 1: one wave32 per segment (index is sorted -> contiguous runs).
//   pass 1: gmax_s = max_i (x_i . Wg + bg)
//   pass 2: e_i = weights_i^pow * exp(gate_i - gmax_s)
//           y[s,:]  = sum_i e_i * x_i   (128 floats, 4 per lane)
//           gs[s]   = sum_i e_i
__global__ void seg_pool_kernel(const float* __restrict__ x,
                                const float* __restrict__ weights,
                                const float* __restrict__ Wg,
                                const float* __restrict__ bg,
                                const float* __restrict__ pw,
                                const int* __restrict__ index,
                                float* __restrict__ y,
                                float* __restrict__ gs,
                                int N, int S) {
    const int lane = threadIdx.x & 31;
    const int wave = threadIdx.x >> 5;
    const int s = blockIdx.x * (blockDim.x >> 5) + wave;
    if (s >= S) return;

    const int start = lower_bound(index, N, s);
    const int end   = lower_bound(index, N, s + 1);

    const float4 wg4 = ((const float4*)Wg)[lane];
    const float bgv = bg[0];
    const float p   = pw[0];

    // pass 1: segment max of gate
    float gmax = -3.402823466e38f;
    for (int i = start; i < end; ++i) {
        __builtin_prefetch(x + (size_t)(i + 1) * DIM, 0, 0);
        float4 r = ((const float4*)(x + (size_t)i * DIM))[lane];
        float d = r.x * wg4.x + r.y * wg4.y + r.z * wg4.z + r.w * wg4.w;
        d = wave_reduce_sum(d) + bgv;       // broadcast to all lanes
        gmax = fmaxf(gmax, d);
    }

    // pass 2: weighted accumulation (rows hot in cache from pass 1)
    float4 acc = make_float4(0.f, 0.f, 0.f, 0.f);
    float gsum = 0.f;
    for (int i = start; i < end; ++i) {
        float4 r = ((const float4*)(x + (size_t)i * DIM))[lane];
        float d = r.x * wg4.x + r.y * wg4.y + r.z * wg4.z + r.w * wg4.w;
        d = wave_reduce_sum(d) + bgv;
        float e = expf(d - gmax) * powf(weights[i], p);
        acc.x += e * r.x; acc.y += e * r.y; acc.z += e * r.z; acc.w += e * r.w;
        gsum += e;
    }

    ((float4*)(y + (size_t)s * DIM))[lane] = acc;
    if (lane == 0) gs[s] = gsum;            // empty segment -> 0 -> out row = 0 (matches ref)
}

// Kernel 2: out[S x 128] = diag(1/(gs+eps)) @ (y @ Wm) + (gs/(gs+eps)) (x) bm
// One wave32 per 16x16 output tile, K-loop of 32 x V_WMMA_F32_16X16X4_F32.
// A (16x4 f32) layout: lanes 0-15 M=lane {K=0,1}; lanes 16-31 M=lane-16 {K=2,3}.
// B (4x16 f32) layout: lanes 0-15 N=lane {K=0,1}; lanes 16-31 N=lane-16 {K=2,3}.
// C/D (16x16 f32): VGPR j -> M = j (lanes 0-15) / j+8 (lanes 16-31), N = lane%16.
//
// Ragged last M-tile: A rows are CLAMPED to S-1 (memory-safe); an out-of-range
// A row only feeds the matching out-of-range D row, which is never stored, so
// no predication is needed around the WMMAs (EXEC stays all-ones, loads are
// unconditional -> clean load clauses, no saveexec per iteration).
// The 1/(gsum+eps) row scaling is applied once in the epilogue, not per K-step.
__global__ void pooled_gemm_wmma_kernel(const float* __restrict__ y,
                                        const float* __restrict__ gs,
                                        const float* __restrict__ Wm,
                                        const float* __restrict__ bm,
                                        float* __restrict__ out,
                                        int S) {
    const int lane = threadIdx.x & 31;
    const int wave = threadIdx.x >> 5;
    const int tile = blockIdx.x * (blockDim.x >> 5) + wave;
    const int Mt = (S + 15) >> 4;
    if (tile >= Mt * 8) return;

    const int row0 = (tile >> 3) << 4;   // M tile origin
    const int n0   = (tile & 7) << 4;    // N tile origin
    const int l16  = lane & 15;
    const int koff = (lane >> 4) << 1;   // 0 for lanes 0-15, 2 for lanes 16-31

    const int arow = min(row0 + l16, S - 1);               // clamped (see note)
    const float* aptr = y  + (size_t)arow * DIM + koff;    // 8B-aligned (koff in {0,2})
    const float* bptr = Wm + (size_t)koff * DIM + (n0 + l16);

    v8f c = {0.f, 0.f, 0.f, 0.f, 0.f, 0.f, 0.f, 0.f};
#pragma unroll
    for (int k = 0; k < DIM; k += 4) {
        v2f a = *(const v2f*)(aptr + k);                   // global_load_b64
        v2f b;
        b[0] = bptr[(size_t)k * DIM];
        b[1] = bptr[(size_t)(k + 1) * DIM];
        c = __builtin_amdgcn_wmma_f32_16x16x4_f32(
                /*neg_a=*/false, a, /*neg_b=*/false, b,
                /*c_mod=*/(short)0, c, /*reuse_a=*/false, /*reuse_b=*/false);
    }

    const float bmv = bm[n0 + l16];
    const int mbase = (lane >> 4) << 3;  // 0 or 8
#pragma unroll
    for (int j = 0; j < 8; ++j) {
        const int r = row0 + mbase + j;
        if (r < S) {
            const float g   = gs[r];
            const float inv = 1.0f / (g + 1e-10f);
            out[(size_t)r * DIM + n0 + l16] = c[j] * inv + (g * inv) * bmv;
        }
    }
}

extern "C" void kernel_launch(void* const* d_in, const int* in_sizes, int n_in,
                              void* d_out, int out_size, void* d_ws, size_t ws_size,
                              hipStream_t stream) {
    const float* x       = (const float*)d_in[0];
    const float* weights = (const float*)d_in[1];
    const float* Wg      = (const float*)d_in[2];
    const float* bg      = (const float*)d_in[3];
    const float* Wm      = (const float*)d_in[4];
    const float* bm      = (const float*)d_in[5];
    const float* pw      = (const float*)d_in[6];
    const int*   index   = (const int*)d_in[7];

    const int N = in_sizes[0] / DIM;   // 1,000,000
    const int S = out_size / DIM;      // 62,500

    float* y  = (float*)d_ws;                 // S*128 floats (pooled features)
    float* gs = y + (size_t)S * DIM;          // S floats (gate sums)

    // Kernel 1: one wave per segment, 8 waves per block
    {
        const int wavesPerBlock = 8;
        dim3 blk(32 * wavesPerBlock);
        dim3 grd((S + wavesPerBlock - 1) / wavesPerBlock);
        seg_pool_kernel<<<grd, blk, 0, stream>>>(x, weights, Wg, bg, pw, index,
                                                 y, gs, N, S);
    }

    // Kernel 2: one wave per 16x16 tile of the S x 128 output
    {
        const int Mt = (S + 15) / 16;
        const int tiles = Mt * 8;              // 128/16 = 8 N-tiles
        const int wavesPerBlock = 8;
        dim3 blk(32 * wavesPerBlock);
        dim3 grd((tiles + wavesPerBlock - 1) / wavesPerBlock);
        pooled_gemm_wmma_kernel<<<grd, blk, 0, stream>>>(y, gs, Wm, bm,
                                                         (float*)d_out, S);
    }
}